// Ernie4_5_MoESparseMoEBlock_12945031430346
// MI455X (gfx1250) — compile-verified
//
#include <hip/hip_runtime.h>
#include <math.h>

// Problem constants (from reference): B=2,S=1024 -> N=2048 tokens
#define N_TOK 2048
#define HDIM 2048
#define NEXP 16
#define IDIM 1024
#define SIDIM 2048
#define CAP 2048        // worst case: every token routes one of its top-2 to same expert

#define LDA 36          // LDS stride for 16x32 A tile (bank-conflict-free WMMA reads)
#define LDB 136         // LDS stride for 32x128 B tile (544B rows: 16B aligned, conflict-free)

typedef float v2f __attribute__((ext_vector_type(2)));
typedef float v8f __attribute__((ext_vector_type(8)));

// -------------------------------------------------- CDNA5 async helpers -----
// Low 32 bits of a generic pointer to __shared__ memory are the LDS byte
// offset (ISA 10.2 aperture mapping).
__device__ __forceinline__ uint32_t lds_addr_of(const void* p) {
  return (uint32_t)(uintptr_t)p;
}
// GVS mode: mem_addr = SADDR(64b, uniform) + VADDR(32b) + IOFFSET. Tracked by
// ASYNCcnt; data goes straight to LDS without touching VGPRs.
__device__ __forceinline__ void async_load_b128(uint32_t lds, const void* base,
                                                uint32_t voff) {
  asm volatile("global_load_async_to_lds_b128 %0, %1, %2"
               :: "v"(lds), "v"(voff), "s"(base) : "memory");
}
__device__ __forceinline__ void async_load_b64(uint32_t lds, const void* base,
                                               uint32_t voff) {
  asm volatile("global_load_async_to_lds_b64 %0, %1, %2"
               :: "v"(lds), "v"(voff), "s"(base) : "memory");
}
__device__ __forceinline__ void wait_async0() {
  asm volatile("s_wait_asynccnt 0" ::: "memory");
}

// ---------------------------------------------------------------- init ------
__global__ __launch_bounds__(32) void moe_init_kernel(int* __restrict__ counts) {
  if (threadIdx.x < NEXP) counts[threadIdx.x] = 0;
}

// -------------------------------------------------------------- router ------
// One wave (32 lanes) per token: fp32 dot products vs all 16 expert rows,
// wave-reduce, lane 0 does softmax/top-2/renorm and appends to expert buckets.
__global__ __launch_bounds__(256) void moe_router_kernel(
    const float* __restrict__ X, const float* __restrict__ RW,
    float* __restrict__ logits_out, int* __restrict__ counts,
    int* __restrict__ assign_tok, float* __restrict__ assign_w) {
  const int lane = threadIdx.x & 31;
  const int wave = threadIdx.x >> 5;
  const int n = blockIdx.x * 8 + wave;
  if (n >= N_TOK) return;

  float acc[NEXP];
#pragma unroll
  for (int e = 0; e < NEXP; ++e) acc[e] = 0.f;
  const float* xr = X + (size_t)n * HDIM;
  for (int h = lane; h < HDIM; h += 32) {
    const float xv = xr[h];
#pragma unroll
    for (int e = 0; e < NEXP; ++e) acc[e] += xv * RW[e * HDIM + h];
  }
#pragma unroll
  for (int off = 16; off > 0; off >>= 1) {
#pragma unroll
    for (int e = 0; e < NEXP; ++e) acc[e] += __shfl_xor(acc[e], off);
  }

  if (lane == 0) {
    float mx = acc[0];
#pragma unroll
    for (int e = 1; e < NEXP; ++e) mx = fmaxf(mx, acc[e]);
    float p[NEXP];
    float sum = 0.f;
#pragma unroll
    for (int e = 0; e < NEXP; ++e) {
      p[e] = __expf(acc[e] - mx);
      sum += p[e];
      logits_out[n * NEXP + e] = acc[e];   // raw logits are the 2nd output
    }
    const float inv = 1.f / sum;
#pragma unroll
    for (int e = 0; e < NEXP; ++e) p[e] *= inv;

    int i0 = 0; float b0 = p[0];
#pragma unroll
    for (int e = 1; e < NEXP; ++e) if (p[e] > b0) { b0 = p[e]; i0 = e; }
    int i1 = (i0 == 0) ? 1 : 0; float b1 = -1.f;
#pragma unroll
    for (int e = 0; e < NEXP; ++e) if (e != i0 && p[e] > b1) { b1 = p[e]; i1 = e; }

    const float s = fmaxf(b0 + b1, 1e-12f);
    const float w0 = b0 / s, w1 = b1 / s;
    const int s0 = atomicAdd(&counts[i0], 1);
    assign_tok[i0 * CAP + s0] = n;
    assign_w[i0 * CAP + s0] = w0;
    const int s1 = atomicAdd(&counts[i1], 1);
    assign_tok[i1 * CAP + s1] = n;
    assign_w[i1 * CAP + s1] = w1;
  }
}

// ------------------------------------------------- padded prefix bases ------
__global__ void moe_prefix_kernel(const int* __restrict__ counts,
                                  int* __restrict__ pbase) {
  if (threadIdx.x == 0 && blockIdx.x == 0) {
    int run = 0;
    for (int e = 0; e < NEXP; ++e) {
      pbase[e] = run;
      run += (counts[e] + 15) & ~15;
    }
    pbase[NEXP] = run;
  }
}

// --------------------------------------------------- fp32 WMMA (K=4) --------
// A 16x4 fp32: lanes 0-15 hold row M=lane, VGPR0=K0,VGPR1=K1; lanes 16-31 K2,K3.
// B 4x16 fp32: VGPR0 = rows {K0 (lanes0-15), K2 (lanes16-31)}, VGPR1 = {K1,K3}.
// C/D 16x16:  VGPR j = rows {j, j+8}, col = lane&15.
__device__ __forceinline__ v8f wmma_f32_k4(const float* __restrict__ As, int arow,
                                           const float* __restrict__ Bs, int col,
                                           int k0, int khalf, v8f acc) {
  v2f a, b;
  a.x = As[arow * LDA + k0 + 2 * khalf];
  a.y = As[arow * LDA + k0 + 2 * khalf + 1];
  b.x = Bs[(k0 + 2 * khalf) * LDB + col];
  b.y = Bs[(k0 + 2 * khalf + 1) * LDB + col];
  return __builtin_amdgcn_wmma_f32_16x16x4_f32(false, a, false, b, (short)0, acc,
                                               false, false);
}

// ----------------------------------------- fused gate/up (SwiGLU front) -----
// C[16 x 128] tile: g = A@Wg, u = A@Wu over K=HDIM, epilogue T = silu(g)*u.
// Double-buffered LDS, filled by global_load_async_to_lds (ASYNCcnt pipeline).
// EXPERT: A rows gathered via assign_tok; padded rows use a clamped (valid)
// token index -> finite junk in padded T rows, skipped by the down epilogue.
template <bool EXPERT>
__global__ __launch_bounds__(256) void moe_gateup_kernel(
    const float* __restrict__ X, const float* __restrict__ Wg_all,
    const float* __restrict__ Wu_all, float* __restrict__ T,
    const int* __restrict__ counts, const int* __restrict__ pbase,
    const int* __restrict__ assign_tok) {
  const int Idim = EXPERT ? IDIM : SIDIM;
  const int bx = blockIdx.x;
  const int colBase = blockIdx.y * 128;
  int e = 0, m16 = bx, cnt = 0, rowBaseT = 0;
  const float *Wg, *Wu;
  if (EXPERT) {
    e = bx >> 7;
    m16 = bx & 127;
    cnt = counts[e];
    if (m16 * 16 >= ((cnt + 15) & ~15)) return;   // uniform early-exit
    rowBaseT = pbase[e] + m16 * 16;
    Wg = Wg_all + (size_t)e * HDIM * IDIM;
    Wu = Wu_all + (size_t)e * HDIM * IDIM;
  } else {
    rowBaseT = m16 * 16;
    Wg = Wg_all;
    Wu = Wu_all;
  }

  __shared__ float As[2][16 * LDA];
  __shared__ float Bg[2][32 * LDB];
  __shared__ float Bu[2][32 * LDB];

  const int tid = threadIdx.x;
  const int lane = tid & 31;
  const int wave = tid >> 5;
  const int col = (wave << 4) | (lane & 15);   // 0..127: this lane's tile column
  const int khalf = lane >> 4;
  const int arow = lane & 15;

  // A-stage mapping: 16 rows x 32 cols, 2 contiguous floats (b64) per thread
  const int ar = tid >> 4;
  const int ac = (tid & 15) * 2;
  int tokRow;
  if (EXPERT) {
    const int slot = m16 * 16 + ar;
    const int cs = slot < cnt ? slot : cnt - 1;  // clamp padded rows to valid
    tokRow = assign_tok[e * CAP + cs];
  } else {
    tokRow = m16 * 16 + ar;
  }
  const uint32_t aVoff0 = (uint32_t)((tokRow * HDIM + ac) * 4);

  // B-stage mapping: 32 rows x 128 cols, 4x b128 per thread per matrix
  uint32_t bVoff0[4], bgLds[2][4], buLds[2][4];
  uint32_t aLds[2];
#pragma unroll
  for (int i = 0; i < 4; ++i) {
    const int fl = tid + i * 256;
    const int br = fl >> 5;
    const int bc = (fl & 31) * 4;
    bVoff0[i] = (uint32_t)((br * Idim + colBase + bc) * 4);
#pragma unroll
    for (int s = 0; s < 2; ++s) {
      bgLds[s][i] = lds_addr_of(&Bg[s][br * LDB + bc]);
      buLds[s][i] = lds_addr_of(&Bu[s][br * LDB + bc]);
    }
  }
#pragma unroll
  for (int s = 0; s < 2; ++s) aLds[s] = lds_addr_of(&As[s][ar * LDA + ac]);

  const uint32_t bRowChunk = (uint32_t)(32 * Idim * 4);  // bytes per K-chunk of B

  auto issue = [&](int s, int kc) {
    async_load_b64(aLds[s], X, aVoff0 + (uint32_t)(kc * 4));
    const uint32_t bk = (uint32_t)(kc >> 5) * bRowChunk;
#pragma unroll
    for (int i = 0; i < 4; ++i) {
      async_load_b128(bgLds[s][i], Wg, bVoff0[i] + bk);
      async_load_b128(buLds[s][i], Wu, bVoff0[i] + bk);
    }
  };

  v8f accG = {0.f, 0.f, 0.f, 0.f, 0.f, 0.f, 0.f, 0.f};
  v8f accU = accG;

  issue(0, 0);
  int s = 0;
  for (int kc = 0; kc < HDIM; kc += 32, s ^= 1) {
    wait_async0();        // our stage-s tile is in LDS
    __syncthreads();      // everyone's stage-s tile is in LDS; stage s^1 reads done
    if (kc + 32 < HDIM) issue(s ^ 1, kc + 32);   // overlap with compute below
#pragma unroll
    for (int k0 = 0; k0 < 32; k0 += 4) {
      accG = wmma_f32_k4(As[s], arow, Bg[s], col, k0, khalf, accG);
      accU = wmma_f32_k4(As[s], arow, Bu[s], col, k0, khalf, accU);
    }
  }

  const int gcol = colBase + col;
#pragma unroll
  for (int j = 0; j < 8; ++j) {
    const int r = j + 8 * khalf;
    const float g = accG[j];
    const float u = accU[j];
    const float t = (g / (1.f + __expf(-g))) * u;   // silu(g)*u
    T[(size_t)(rowBaseT + r) * Idim + gcol] = t;
  }
}

// ------------------------------------------------ down projection -----------
// EXPERT: out[tok,:] += w * (T @ Wd) via global_atomic_add_f32 scatter.
// Shared: out written directly (it is the first writer of d_out).
template <bool EXPERT>
__global__ __launch_bounds__(256) void moe_down_kernel(
    const float* __restrict__ T, const float* __restrict__ Wd_all,
    float* __restrict__ out, const int* __restrict__ counts,
    const int* __restrict__ pbase, const int* __restrict__ assign_tok,
    const float* __restrict__ assign_w) {
  const int Kd = EXPERT ? IDIM : SIDIM;
  const int bx = blockIdx.x;
  const int colBase = blockIdx.y * 128;
  int e = 0, m16 = bx, cnt = 0, rowBaseT = 0;
  const float* Wd;
  if (EXPERT) {
    e = bx >> 7;
    m16 = bx & 127;
    cnt = counts[e];
    if (m16 * 16 >= ((cnt + 15) & ~15)) return;
    rowBaseT = pbase[e] + m16 * 16;
    Wd = Wd_all + (size_t)e * IDIM * HDIM;
  } else {
    rowBaseT = m16 * 16;
    Wd = Wd_all;
  }

  __shared__ float As[2][16 * LDA];
  __shared__ float Bs[2][32 * LDB];

  const int tid = threadIdx.x;
  const int lane = tid & 31;
  const int wave = tid >> 5;
  const int col = (wave << 4) | (lane & 15);
  const int khalf = lane >> 4;
  const int arow = lane & 15;
  const int ar = tid >> 4;
  const int ac = (tid & 15) * 2;

  const uint32_t aVoff0 = (uint32_t)(((rowBaseT + ar) * Kd + ac) * 4);
  uint32_t bVoff0[4], bLds[2][4], aLds[2];
#pragma unroll
  for (int i = 0; i < 4; ++i) {
    const int fl = tid + i * 256;
    const int br = fl >> 5;
    const int bc = (fl & 31) * 4;
    bVoff0[i] = (uint32_t)((br * HDIM + colBase + bc) * 4);
#pragma unroll
    for (int s = 0; s < 2; ++s) bLds[s][i] = lds_addr_of(&Bs[s][br * LDB + bc]);
  }
#pragma unroll
  for (int s = 0; s < 2; ++s) aLds[s] = lds_addr_of(&As[s][ar * LDA + ac]);

  const uint32_t bRowChunk = (uint32_t)(32 * HDIM * 4);

  auto issue = [&](int s, int kc) {
    async_load_b64(aLds[s], T, aVoff0 + (uint32_t)(kc * 4));
    const uint32_t bk = (uint32_t)(kc >> 5) * bRowChunk;
#pragma unroll
    for (int i = 0; i < 4; ++i) async_load_b128(bLds[s][i], Wd, bVoff0[i] + bk);
  };

  v8f acc = {0.f, 0.f, 0.f, 0.f, 0.f, 0.f, 0.f, 0.f};

  issue(0, 0);
  int s = 0;
  for (int kc = 0; kc < Kd; kc += 32, s ^= 1) {
    wait_async0();
    __syncthreads();
    if (kc + 32 < Kd) issue(s ^ 1, kc + 32);
#pragma unroll
    for (int k0 = 0; k0 < 32; k0 += 4)
      acc = wmma_f32_k4(As[s], arow, Bs[s], col, k0, khalf, acc);
  }

  const int gcol = colBase + col;
#pragma unroll
  for (int j = 0; j < 8; ++j) {
    const int r = j + 8 * khalf;
    if (EXPERT) {
      const int slot = m16 * 16 + r;
      if (slot < cnt) {
        const int tok = assign_tok[e * CAP + slot];
        const float w = assign_w[e * CAP + slot];
        unsafeAtomicAdd(&out[(size_t)tok * HDIM + gcol], w * acc[j]);
      }
    } else {
      out[(size_t)(rowBaseT + r) * HDIM + gcol] = acc[j];
    }
  }
}

// ---------------------------------------------------------------- host ------
extern "C" void kernel_launch(void* const* d_in, const int* in_sizes, int n_in,
                              void* d_out, int out_size, void* d_ws,
                              size_t ws_size, hipStream_t stream) {
  (void)in_sizes; (void)n_in; (void)out_size; (void)ws_size;
  const float* X  = (const float*)d_in[0];   // [N, H]
  const float* RW = (const float*)d_in[1];   // [E, H]
  const float* Wg = (const float*)d_in[2];   // [E, H, I]
  const float* Wu = (const float*)d_in[3];   // [E, H, I]
  const float* Wd = (const float*)d_in[4];   // [E, I, H]
  const float* SG = (const float*)d_in[5];   // [H, SI]
  const float* SU = (const float*)d_in[6];   // [H, SI]
  const float* SD = (const float*)d_in[7];   // [SI, H]

  float* out = (float*)d_out;                        // [N, H]
  float* logits = out + (size_t)N_TOK * HDIM;        // [N, E]

  // Workspace layout (~18 MB):
  char* ws = (char*)d_ws;
  int* counts = (int*)ws;                                    // 16 ints
  int* pbase = (int*)(ws + 64);                              // 17 ints
  int* assign_tok = (int*)(ws + 256);                        // E*CAP ints
  float* assign_w = (float*)(ws + 256 + NEXP * CAP * 4);     // E*CAP floats
  float* T = (float*)(ws + 256 + 2 * NEXP * CAP * 4);        // up to 4336*1024 f32

  moe_init_kernel<<<1, 32, 0, stream>>>(counts);
  moe_router_kernel<<<N_TOK / 8, 256, 0, stream>>>(X, RW, logits, counts,
                                                   assign_tok, assign_w);
  moe_prefix_kernel<<<1, 1, 0, stream>>>(counts, pbase);

  // Shared expert: T = silu(X@SG)*(X@SU); out = T@SD (first writer of out)
  moe_gateup_kernel<false><<<dim3(N_TOK / 16, SIDIM / 128), 256, 0, stream>>>(
      X, SG, SU, T, counts, pbase, assign_tok);
  moe_down_kernel<false><<<dim3(N_TOK / 16, HDIM / 128), 256, 0, stream>>>(
      T, SD, out, counts, pbase, assign_tok, assign_w);

  // Sparse experts (top-2 gathered): T = silu(Xg@Wg)*(Xg@Wu); out += w*(T@Wd)
  moe_gateup_kernel<true><<<dim3(NEXP * (CAP / 16), IDIM / 128), 256, 0, stream>>>(
      X, Wg, Wu, T, counts, pbase, assign_tok);
  moe_down_kernel<true><<<dim3(NEXP * (CAP / 16), HDIM / 128), 256, 0, stream>>>(
      T, Wd, out, counts, pbase, assign_tok, assign_w);
}